// GATModel_5368709120236
// MI455X (gfx1250) — compile-verified
//
#include <hip/hip_runtime.h>
#include <math.h>

// ---------------------------------------------------------------------------
// GAT 2-layer forward for MI455X (gfx1250, wave32, WMMA)
// ---------------------------------------------------------------------------

#define N_NODES 50000
#define N_EDGES0 500000
#define N_EDGES (N_EDGES0 + N_NODES)   // + self loops = 550000
#define IN_DIM 128
#define HIDDEN 32
#define HEADS 8
#define OUT_DIM 256                    // HIDDEN*HEADS
#define NEG_SLOPE 0.2f

typedef float v2f __attribute__((ext_vector_type(2)));
typedef float v8f __attribute__((ext_vector_type(8)));

// ---------------- edge list with self loops -------------------------------
__global__ void build_edges(const int* __restrict__ ei, int* __restrict__ src,
                            int* __restrict__ dst) {
  int i = blockIdx.x * blockDim.x + threadIdx.x;
  if (i < N_EDGES0) {
    src[i] = ei[i];
    dst[i] = ei[N_EDGES0 + i];
  } else if (i < N_EDGES) {
    int n = i - N_EDGES0;
    src[i] = n;
    dst[i] = n;
  }
}

// ---------------- generic fill --------------------------------------------
__global__ void fill_f32(float* __restrict__ p, float v, int n) {
  int i = blockIdx.x * blockDim.x + threadIdx.x;
  if (i < n) p[i] = v;
}

// ---------------- fp32 WMMA GEMM: C[M,Nc] = A[M,K] @ B[K,Nc] --------------
// One wave computes a 16x64 strip: 4 independent 16x16 accumulators fed by a
// shared A fragment -> 4 hazard-free V_WMMA_F32_16X16X4_F32 per k-step.
// M multiple of 16 (50000 = 3125*16), K multiple of 4, Nc multiple of 64.
__global__ __launch_bounds__(256) void gemm_wmma_f32(
    const float* __restrict__ A, const float* __restrict__ B,
    float* __restrict__ C, int M, int K, int Nc) {
  int wave = (blockIdx.x * blockDim.x + threadIdx.x) >> 5;
  int lane = threadIdx.x & 31;
  int colGroups = Nc >> 6;                  // 64-wide column groups
  int tileRow = wave / colGroups;
  int grp = wave - tileRow * colGroups;
  if (tileRow * 16 >= M) return;            // wave-uniform guard (EXEC stays full)

  int row0 = tileRow << 4;
  int col0 = grp << 6;
  int l16 = lane & 15;
  int hi  = lane >> 4;                      // 0: K=k,k+1   1: K=k+2,k+3

  // A fragment: lane holds A[row0+l16][k + 2*hi + {0,1}] (one aligned v2f load)
  const float* aptr = A + (size_t)(row0 + l16) * K + 2 * hi;
  // B fragments: lane holds B[k + 2*hi + {0,1}][col0 + 16*t + l16], t=0..3
  const float* bbase = B + (size_t)(2 * hi) * Nc + col0 + l16;

  v8f acc0 = {}, acc1 = {}, acc2 = {}, acc3 = {};
  for (int k = 0; k < K; k += 4) {
    v2f a = *(const v2f*)(aptr + k);
    const float* bp = bbase + (size_t)k * Nc;
    v2f b0, b1, b2, b3;
    b0.x = bp[0];  b0.y = bp[Nc];
    b1.x = bp[16]; b1.y = bp[Nc + 16];
    b2.x = bp[32]; b2.y = bp[Nc + 32];
    b3.x = bp[48]; b3.y = bp[Nc + 48];
    acc0 = __builtin_amdgcn_wmma_f32_16x16x4_f32(false, a, false, b0, (short)0, acc0, false, false);
    acc1 = __builtin_amdgcn_wmma_f32_16x16x4_f32(false, a, false, b1, (short)0, acc1, false, false);
    acc2 = __builtin_amdgcn_wmma_f32_16x16x4_f32(false, a, false, b2, (short)0, acc2, false, false);
    acc3 = __builtin_amdgcn_wmma_f32_16x16x4_f32(false, a, false, b3, (short)0, acc3, false, false);
  }

  // C/D layout: VGPR v -> M=row0+v (lanes 0-15) / row0+v+8 (lanes 16-31), N=l16
  float* cp = C + (size_t)(row0 + 8 * hi) * Nc + col0 + l16;
#pragma unroll
  for (int v = 0; v < 8; ++v) {
    float* r = cp + (size_t)v * Nc;
    r[0]  = acc0[v];
    r[16] = acc1[v];
    r[32] = acc2[v];
    r[48] = acc3[v];
  }
}

// ---------------- per-node attention logits -------------------------------
// e_src[n,h] = sum_c h[n,h,c]*a_src[h,c];  e_dst likewise
__global__ void node_e(const float* __restrict__ h, const float* __restrict__ a_src,
                       const float* __restrict__ a_dst, float* __restrict__ e_src,
                       float* __restrict__ e_dst, int N, int H, int Cc) {
  int i = blockIdx.x * blockDim.x + threadIdx.x;
  if (i >= N * H) return;
  int n = i / H;
  int hh = i - n * H;
  const float* hp = h + (size_t)n * H * Cc + (size_t)hh * Cc;
  const float* as = a_src + (size_t)hh * Cc;
  const float* ad = a_dst + (size_t)hh * Cc;
  float s = 0.f, d = 0.f;
  for (int c = 0; c < Cc; ++c) {
    float v = hp[c];
    s += v * as[c];
    d += v * ad[c];
  }
  e_src[i] = s;
  e_dst[i] = d;
}

// ---------------- float atomic max via int punning ------------------------
__device__ inline void atomicMaxF(float* addr, float val) {
  if (val >= 0.f)
    atomicMax((int*)addr, __float_as_int(val));
  else
    atomicMin((unsigned int*)addr, (unsigned int)__float_as_int(val));
}

__device__ inline float lrelu(float v) { return v > 0.f ? v : NEG_SLOPE * v; }

// ---------------- pass 1: segment max of edge logits ----------------------
__global__ void edge_max(const int* __restrict__ src, const int* __restrict__ dst,
                         const float* __restrict__ e_src, const float* __restrict__ e_dst,
                         float* __restrict__ m, int E, int H) {
  int i = blockIdx.x * blockDim.x + threadIdx.x;
  if (i >= E * H) return;
  int e = i / H;
  int hh = i - e * H;
  int s = src[e], d = dst[e];
  float v = lrelu(e_src[s * H + hh] + e_dst[d * H + hh]);
  atomicMaxF(&m[d * H + hh], v);
}

// ---------------- pass 2: exp(e - m) and segment sum ----------------------
__global__ void edge_expsum(const int* __restrict__ src, const int* __restrict__ dst,
                            const float* __restrict__ e_src, const float* __restrict__ e_dst,
                            const float* __restrict__ m, float* __restrict__ eexp,
                            float* __restrict__ denom, int E, int H) {
  int i = blockIdx.x * blockDim.x + threadIdx.x;
  if (i >= E * H) return;
  int e = i / H;
  int hh = i - e * H;
  int s = src[e], d = dst[e];
  float v = lrelu(e_src[s * H + hh] + e_dst[d * H + hh]);
  float ee = expf(v - m[d * H + hh]);
  eexp[i] = ee;
  atomicAdd(&denom[d * H + hh], ee);
}

// ---------------- pass 3: weighted message scatter (float4) ---------------
// out[dst, c..c+3] += h[src, c..c+3] * alpha(edge, head(c))
__global__ void edge_scatter(const int* __restrict__ src, const int* __restrict__ dst,
                             const float* __restrict__ eexp, const float* __restrict__ denom,
                             const float* __restrict__ h, float* __restrict__ out,
                             int E, int H, int Cc) {
  long long i = (long long)blockIdx.x * blockDim.x + threadIdx.x;
  int D = H * Cc;
  int Dq = D >> 2;                               // float4 groups per edge
  if (i >= (long long)E * Dq) return;
  int e = (int)(i / Dq);
  int q = (int)(i - (long long)e * Dq);
  int c = q << 2;
  int hh = c / Cc;                               // Cc multiple of 4 -> uniform in group
  int s = src[e], d = dst[e];
  float alpha = eexp[(size_t)e * H + hh] / (denom[d * H + hh] + 1e-16f);
  const float4 hv = *(const float4*)(h + (size_t)s * D + c);
  float* op = out + (size_t)d * D + c;
  atomicAdd(op + 0, hv.x * alpha);
  atomicAdd(op + 1, hv.y * alpha);
  atomicAdd(op + 2, hv.z * alpha);
  atomicAdd(op + 3, hv.w * alpha);
}

// ---------------- bias + ELU (in place) -----------------------------------
__global__ void bias_elu(float* __restrict__ x, const float* __restrict__ b, int N, int D) {
  int i = blockIdx.x * blockDim.x + threadIdx.x;
  if (i >= N * D) return;
  int c = i % D;
  float v = x[i] + b[c];
  x[i] = v > 0.f ? v : (expf(v) - 1.f);
}

// ---------------- bias add (in place) -------------------------------------
__global__ void bias_add(float* __restrict__ x, const float* __restrict__ b, int N, int D) {
  int i = blockIdx.x * blockDim.x + threadIdx.x;
  if (i >= N * D) return;
  x[i] += b[i % D];
}

// ---------------------------------------------------------------------------
extern "C" void kernel_launch(void* const* d_in, const int* in_sizes, int n_in,
                              void* d_out, int out_size, void* d_ws, size_t ws_size,
                              hipStream_t stream) {
  const float* x      = (const float*)d_in[0];
  const int*   ei     = (const int*)  d_in[1];
  const float* W1     = (const float*)d_in[2];
  const float* a_src1 = (const float*)d_in[3];
  const float* a_dst1 = (const float*)d_in[4];
  const float* b1     = (const float*)d_in[5];
  const float* W2     = (const float*)d_in[6];
  const float* a_src2 = (const float*)d_in[7];
  const float* a_dst2 = (const float*)d_in[8];
  const float* b2     = (const float*)d_in[9];
  float* out = (float*)d_out;

  // ---- workspace carve-up (aligned to 256B) ----
  char* ws = (char*)d_ws;
  size_t off = 0;
  auto take = [&](size_t bytes) -> void* {
    void* p = ws + off;
    off = (off + bytes + 255) & ~(size_t)255;
    return p;
  };
  int*   src_all = (int*)  take((size_t)N_EDGES * 4);
  int*   dst_all = (int*)  take((size_t)N_EDGES * 4);
  float* hbuf    = (float*)take((size_t)N_NODES * OUT_DIM * 4);  // h1, later h2
  float* o1      = (float*)take((size_t)N_NODES * OUT_DIM * 4);  // layer1 out / gemm2 in
  float* e_src   = (float*)take((size_t)N_NODES * HEADS * 4);
  float* e_dst   = (float*)take((size_t)N_NODES * HEADS * 4);
  float* m       = (float*)take((size_t)N_NODES * HEADS * 4);
  float* denom   = (float*)take((size_t)N_NODES * HEADS * 4);
  float* eexp    = (float*)take((size_t)N_EDGES * HEADS * 4);
  (void)ws_size; (void)in_sizes; (void)n_in; (void)out_size;

  const int T = 256;
  auto blk = [](long long n, int t) { return (unsigned)((n + t - 1) / t); };

  // ---- edge list with self loops ----
  build_edges<<<blk(N_EDGES, T), T, 0, stream>>>(ei, src_all, dst_all);

  // ======================= Layer 1 =======================
  // h1 = x @ W1   (50000x128 @ 128x256) -- WMMA fp32, 16x64 per wave
  {
    long long waves = (long long)(N_NODES / 16) * (OUT_DIM / 64);  // 3125*4
    gemm_wmma_f32<<<blk(waves * 32, T), T, 0, stream>>>(
        x, W1, hbuf, N_NODES, IN_DIM, OUT_DIM);
  }

  node_e<<<blk((long long)N_NODES * HEADS, T), T, 0, stream>>>(
      hbuf, a_src1, a_dst1, e_src, e_dst, N_NODES, HEADS, HIDDEN);

  fill_f32<<<blk((long long)N_NODES * HEADS, T), T, 0, stream>>>(m, -INFINITY, N_NODES * HEADS);
  fill_f32<<<blk((long long)N_NODES * HEADS, T), T, 0, stream>>>(denom, 0.f, N_NODES * HEADS);
  fill_f32<<<blk((long long)N_NODES * OUT_DIM, T), T, 0, stream>>>(o1, 0.f, N_NODES * OUT_DIM);

  edge_max<<<blk((long long)N_EDGES * HEADS, T), T, 0, stream>>>(
      src_all, dst_all, e_src, e_dst, m, N_EDGES, HEADS);
  edge_expsum<<<blk((long long)N_EDGES * HEADS, T), T, 0, stream>>>(
      src_all, dst_all, e_src, e_dst, m, eexp, denom, N_EDGES, HEADS);
  edge_scatter<<<blk((long long)N_EDGES * (OUT_DIM / 4), T), T, 0, stream>>>(
      src_all, dst_all, eexp, denom, hbuf, o1, N_EDGES, HEADS, HIDDEN);

  // out1 += b1 ; ELU  (in place, becomes layer-2 input)
  bias_elu<<<blk((long long)N_NODES * OUT_DIM, T), T, 0, stream>>>(o1, b1, N_NODES, OUT_DIM);

  // ======================= Layer 2 =======================
  // h2 = elu(out1) @ W2   (50000x256 @ 256x256) -- WMMA fp32, reuse hbuf
  {
    long long waves = (long long)(N_NODES / 16) * (OUT_DIM / 64);
    gemm_wmma_f32<<<blk(waves * 32, T), T, 0, stream>>>(
        o1, W2, hbuf, N_NODES, OUT_DIM, OUT_DIM);
  }

  node_e<<<blk((long long)N_NODES * 1, T), T, 0, stream>>>(
      hbuf, a_src2, a_dst2, e_src, e_dst, N_NODES, 1, OUT_DIM);

  fill_f32<<<blk((long long)N_NODES, T), T, 0, stream>>>(m, -INFINITY, N_NODES);
  fill_f32<<<blk((long long)N_NODES, T), T, 0, stream>>>(denom, 0.f, N_NODES);
  fill_f32<<<blk((long long)N_NODES * OUT_DIM, T), T, 0, stream>>>(out, 0.f, N_NODES * OUT_DIM);

  edge_max<<<blk((long long)N_EDGES * 1, T), T, 0, stream>>>(
      src_all, dst_all, e_src, e_dst, m, N_EDGES, 1);
  edge_expsum<<<blk((long long)N_EDGES * 1, T), T, 0, stream>>>(
      src_all, dst_all, e_src, e_dst, m, eexp, denom, N_EDGES, 1);
  edge_scatter<<<blk((long long)N_EDGES * (OUT_DIM / 4), T), T, 0, stream>>>(
      src_all, dst_all, eexp, denom, hbuf, out, N_EDGES, 1, OUT_DIM);

  // heads=1, concat=False -> mean over 1 head is identity; just add bias
  bias_add<<<blk((long long)N_NODES * OUT_DIM, T), T, 0, stream>>>(out, b2, N_NODES, OUT_DIM);
}